// SimpleMFWithPropagation_47425028882648
// MI455X (gfx1250) — compile-verified
//
#include <hip/hip_runtime.h>

typedef __attribute__((ext_vector_type(2))) float v2f;
typedef __attribute__((ext_vector_type(8))) float v8f;

#define EMB_DIM 64

// ---------------- Kernel 1: zero the propagation accumulator ----------------
__global__ __launch_bounds__(256) void zero_ws_kernel(float4* ws, int n4) {
    int t = blockIdx.x * blockDim.x + threadIdx.x;
    if (t < n4) ws[t] = make_float4(0.f, 0.f, 0.f, 0.f);
}

// ---------------- Kernel 2: gather-scale-scatter (segment sum) --------------
// One wave (32 lanes) per edge: edge metadata is wave-uniform (scalarizable),
// each lane handles 2 consecutive dims -> float2 load + 2x global_atomic_add_f32.
// Working set (emb 38.4MB + accum 38.4MB) fits in the 192MB L2, so the random
// traffic stays on-die; HBM only streams the edge lists.
__global__ __launch_bounds__(256) void edge_scatter_kernel(
    const float* __restrict__ uemb, const float* __restrict__ iemb,
    const int* __restrict__ rows, const int* __restrict__ cols,
    const float* __restrict__ vals, float* __restrict__ prop,
    int nnz, int n_users) {
    int t = blockIdx.x * blockDim.x + threadIdx.x;
    int e = t >> 5;                 // wave-uniform edge id
    if (e >= nnz) return;
    int p = (t & 31) << 1;          // dim pair handled by this lane
    int col = cols[e];
    int row = rows[e];
    float val = vals[e];
    const float* src = (col < n_users)
                           ? (uemb + (size_t)col * EMB_DIM)
                           : (iemb + (size_t)(col - n_users) * EMB_DIM);
    float2 m = *(const float2*)(src + p);
    float* dst = prop + (size_t)row * EMB_DIM + p;
    __hip_atomic_fetch_add(dst,     val * m.x, __ATOMIC_RELAXED, __HIP_MEMORY_SCOPE_AGENT);
    __hip_atomic_fetch_add(dst + 1, val * m.y, __ATOMIC_RELAXED, __HIP_MEMORY_SCOPE_AGENT);
}

// ---------------- Kernel 3: batched dot products via WMMA f32 ---------------
// One wave scores 16 (user,item) pairs. A = 16 user rows (16x4 K-chunk),
// B = 16 item rows as 4x16; accumulate D = A*B + C over K=64 (16 wmma steps);
// dot products are the diagonal of the final 16x16 f32 accumulator.
// A-operand layout (16x4 f32): lane half selects K base {0,2}; each lane holds
// a contiguous float2 {K=kb, K=kb+1} of its row -> plain b64 loads.
__global__ __launch_bounds__(256) void score_wmma_kernel(
    const float* __restrict__ uprop, const float* __restrict__ iprop,
    const int* __restrict__ uidx, const int* __restrict__ iidx,
    float* __restrict__ out, int ngroups) {
    int lane = threadIdx.x & 31;
    int wave = (blockIdx.x * blockDim.x + threadIdx.x) >> 5;
    if (wave >= ngroups) return;                 // wave-uniform branch

    int j  = wave * 16 + (lane & 15);            // batch element this lane feeds
    int kb = (lane >> 4) << 1;                   // 0 or 2: K sub-base per lane half
    const float* urow = uprop + (size_t)uidx[j] * EMB_DIM + kb;
    const float* irow = iprop + (size_t)iidx[j] * EMB_DIM + kb;

    v8f c = {};
#pragma unroll
    for (int k0 = 0; k0 < EMB_DIM; k0 += 4) {
        v2f a = *(const v2f*)(urow + k0);        // A: M=lane&15, K=k0+kb..+1
        v2f b = *(const v2f*)(irow + k0);        // B: N=lane&15, K=k0+kb..+1
        c = __builtin_amdgcn_wmma_f32_16x16x4_f32(
                /*neg_a=*/false, a, /*neg_b=*/false, b,
                /*c_mod=*/(short)0, c, /*reuse_a=*/false, /*reuse_b=*/false);
    }

    // Diagonal extraction. D layout: VGPR r, lanes 0-15 -> M=r, N=lane;
    // lanes 16-31 -> M=r+8, N=lane-16. Diagonal M=N=m lives at:
    //   m in 0..7  : vgpr m,   lane m       (lane>>3 == 0)
    //   m in 8..15 : vgpr m-8, lane m+16    (lane>>3 == 3)
    int grp = lane >> 3;
    if (grp == 0 || grp == 3) {
        int r = lane & 7;
        float d = c[0];
#pragma unroll
        for (int q = 1; q < 8; ++q) d = (r == q) ? c[q] : d;
        out[wave * 16 + (lane & 15)] = d;
    }
}

// ---------------- Scalar tail (only if batch % 16 != 0; unused for 16384) ---
__global__ __launch_bounds__(64) void score_tail_kernel(
    const float* __restrict__ uprop, const float* __restrict__ iprop,
    const int* __restrict__ uidx, const int* __restrict__ iidx,
    float* __restrict__ out, int start, int batch) {
    int b = start + blockIdx.x * blockDim.x + threadIdx.x;
    if (b >= batch) return;
    const float* ur = uprop + (size_t)uidx[b] * EMB_DIM;
    const float* ir = iprop + (size_t)iidx[b] * EMB_DIM;
    float acc = 0.f;
#pragma unroll
    for (int k = 0; k < EMB_DIM; ++k) acc = fmaf(ur[k], ir[k], acc);
    out[b] = acc;
}

extern "C" void kernel_launch(void* const* d_in, const int* in_sizes, int n_in,
                              void* d_out, int out_size, void* d_ws, size_t ws_size,
                              hipStream_t stream) {
    const float* uemb = (const float*)d_in[0];
    const float* iemb = (const float*)d_in[1];
    const int*   rows = (const int*)d_in[2];
    const int*   cols = (const int*)d_in[3];
    const float* vals = (const float*)d_in[4];
    const int*   u    = (const int*)d_in[5];
    const int*   iv   = (const int*)d_in[6];

    int n_users = in_sizes[0] / EMB_DIM;   // 100000
    int n_items = in_sizes[1] / EMB_DIM;   // 50000
    int nnz     = in_sizes[2];             // 1200000
    int batch   = in_sizes[5];             // 16384
    int n_nodes = n_users + n_items;

    float* prop = (float*)d_ws;            // 150000 x 64 f32 accumulator (38.4 MB)

    // 1) zero accumulator
    int n4 = (n_nodes * EMB_DIM) / 4;
    zero_ws_kernel<<<(n4 + 255) / 256, 256, 0, stream>>>((float4*)d_ws, n4);

    // 2) scatter-add propagation: one wave per edge
    long long tot = (long long)nnz * 32;
    int blocks = (int)((tot + 255) / 256);
    edge_scatter_kernel<<<blocks, 256, 0, stream>>>(uemb, iemb, rows, cols, vals,
                                                    prop, nnz, n_users);

    // 3) WMMA scoring: 16 dot products per wave
    const float* uprop = prop;
    const float* iprop = prop + (size_t)n_users * EMB_DIM;
    int ngroups = batch / 16;
    if (ngroups > 0) {
        int sblocks = (ngroups * 32 + 255) / 256;
        score_wmma_kernel<<<sblocks, 256, 0, stream>>>(uprop, iprop, u, iv,
                                                       (float*)d_out, ngroups);
    }
    int rem = batch - ngroups * 16;
    if (rem > 0) {
        score_tail_kernel<<<(rem + 63) / 64, 64, 0, stream>>>(
            uprop, iprop, u, iv, (float*)d_out, ngroups * 16, batch);
    }
}